// PostprocessLayer_34127810134746
// MI455X (gfx1250) — compile-verified
//
#include <hip/hip_runtime.h>
#include <stdint.h>

#define GH 52
#define GW 52
#define NB 5
#define NBOX (GH * GW * NB)     /* 13520 */
#define NPAD 16384              /* next pow2 >= NBOX */
#define NWPAD 424               /* 53 col-tiles * 8 words, >= ceil(13520/32)=423 */
#define IOU_TH 0.4f
#define SX 8.0f                 /* 416/52 */
#define SY 8.0f
#define NORM_H 416.0f

extern __shared__ __align__(16) unsigned char dynsmem[];

typedef unsigned int v4u __attribute__((ext_vector_type(4)));
typedef int          v8i __attribute__((ext_vector_type(8)));
typedef int          v4i __attribute__((ext_vector_type(4)));

// ---------------- CDNA5 async global->LDS staging (ASYNCcnt) ----------------
__device__ __forceinline__ void async_ld_b128(void* lds_ptr, const void* gptr) {
    // VDST = LDS byte address (low 32 bits of generic LDS pointer),
    // VADDR = 64-bit global address, GV mode (saddr = off).
    unsigned lds_addr = (unsigned)(unsigned long long)lds_ptr;
    unsigned long long gaddr = (unsigned long long)gptr;
    asm volatile("global_load_async_to_lds_b128 %0, %1, off"
                 :: "v"(lds_addr), "v"(gaddr) : "memory");
}

__device__ __forceinline__ void wait_async0() {
#if __has_builtin(__builtin_amdgcn_s_wait_asynccnt)
    __builtin_amdgcn_s_wait_asynccnt(0);
#else
    asm volatile("s_wait_asynccnt 0" ::: "memory");
#endif
}

__device__ __forceinline__ void wait_tensor0() {
#if __has_builtin(__builtin_amdgcn_s_wait_tensorcnt)
    __builtin_amdgcn_s_wait_tensorcnt(0);
#else
    asm volatile("s_wait_tensorcnt 0" ::: "memory");
#endif
}

// ---------------- CDNA5 Tensor Data Mover: 1-D contiguous global->LDS DMA ---
// Copies nelem8 * 8 bytes from gptr to lds_ptr with one TENSOR_LOAD_TO_LDS.
// D# per CDNA5 ISA: group0 = {count=1 | lds_addr |
// global_addr | type=2}; group1 = {data_size=3 (8B), tensor_dim0=tile_dim0=
// nelem8, tensor_dim1=1, tile_dim1=1, stride0=nelem8}; groups 2/3 unused (zero).
__device__ __forceinline__ void tdm_load_1d_to_lds(void* lds_ptr, const void* gptr,
                                                   unsigned nelem8) {
#if __has_builtin(__builtin_amdgcn_tensor_load_to_lds)
    unsigned lds_addr = (unsigned)(unsigned long long)lds_ptr;
    unsigned long long ga = (unsigned long long)gptr;
    v4u g0 = { 1u,                                                   // count=1 (user)
               lds_addr,                                             // bits 63:32
               (unsigned)(ga & 0xFFFFFFFFull),                       // gaddr lo
               (unsigned)(((ga >> 32) & 0x01FFFFFFull) | 0x80000000u) }; // gaddr hi | type=2
    v8i g1 = { (int)(3u << 16),                                      // data_size=8B, wg_mask=0
               (int)((nelem8 & 0xFFFFu) << 16),                      // tensor_dim0[15:0]
               (int)(((nelem8 >> 16) & 0xFFFFu) | (1u << 16)),       // dim0[31:16] | tensor_dim1=1
               (int)((nelem8 & 0xFFFFu) << 16),                      // tensor_dim1 hi=0 | tile_dim0
               (int)1,                                               // tile_dim1=1, tile_dim2=0
               (int)nelem8,                                          // tensor_dim0_stride[31:0]
               0, 0 };
    v4i gz = { 0, 0, 0, 0 };
#if __clang_major__ >= 23
    v8i gz8 = { 0, 0, 0, 0, 0, 0, 0, 0 };
    __builtin_amdgcn_tensor_load_to_lds(g0, g1, gz, gz, gz8, 0);
#else
    __builtin_amdgcn_tensor_load_to_lds(g0, g1, gz, gz, 0);
#endif
#else
    (void)lds_ptr; (void)gptr; (void)nelem8;
#endif
}

__device__ __forceinline__ float iou(float4 a, float4 b) {
    float areaA = fmaxf(a.z - a.x, 0.f) * fmaxf(a.w - a.y, 0.f);
    float areaB = fmaxf(b.z - b.x, 0.f) * fmaxf(b.w - b.y, 0.f);
    float iw = fmaxf(fminf(a.z, b.z) - fmaxf(a.x, b.x), 0.f);
    float ih = fmaxf(fminf(a.w, b.w) - fmaxf(a.y, b.y), 0.f);
    float inter = iw * ih;
    float uni = areaA + areaB - inter;
    return (uni > 0.f) ? inter / fmaxf(uni, 1e-12f) : 0.f;
}

// ---------------- 1) decode boxes + build sort keys ----------------
__global__ void decode_kernel(const float* __restrict__ x, float* __restrict__ out,
                              float4* __restrict__ box, float* __restrict__ score,
                              unsigned long long* __restrict__ keys) {
    int n = blockIdx.x * blockDim.x + threadIdx.x;
    if (n < NBOX) {
        int cell = n / NB;          // gy*GW + gx
        int gx = cell % GW;
        int gy = cell / GW;
        const float* p = x + n * 5;
        float tx = p[0], ty = p[1], tw = p[2], th = p[3], sc = p[4];
        float cx = (tx + (float)gx) * SX;
        float w  = tw * SX;
        float cy = NORM_H - (ty + (float)gy) * SY;   // y-flip per reference
        float h  = th * SY;
        float x1 = cx - 0.5f * w, y1 = cy - 0.5f * h;
        float x2 = cx + 0.5f * w, y2 = cy + 0.5f * h;
        out[n * 5 + 0] = x1; out[n * 5 + 1] = y1;
        out[n * 5 + 2] = x2; out[n * 5 + 3] = y2;
        box[n] = make_float4(x1, y1, x2, y2);
        score[n] = sc;
        // monotonic float->uint, then invert for descending; idx in low bits
        unsigned u = __float_as_uint(sc);
        u = (u & 0x80000000u) ? ~u : (u | 0x80000000u);
        unsigned k = ~u;
        keys[n] = ((unsigned long long)k << 32) | (unsigned)n;
    } else if (n < NPAD) {
        keys[n] = 0xFFFFFFFFFFFFFFFFull;   // sorts to the end
    }
}

// ---------------- 2) single-block bitonic sort, entirely in LDS (128 KB) -----
__global__ void sort_kernel(unsigned long long* __restrict__ keys) {
    unsigned long long* skey = (unsigned long long*)dynsmem;
#if __has_builtin(__builtin_amdgcn_tensor_load_to_lds)
    // One TDM DMA (128 KB contiguous) issued by wave 0; TENSORcnt tracks it.
    if (threadIdx.x < 32) {
        tdm_load_1d_to_lds(skey, keys, NPAD);
        wait_tensor0();
    }
#else
    for (int i = threadIdx.x; i < NPAD / 2; i += blockDim.x)
        async_ld_b128(&skey[i * 2], &keys[i * 2]);
    wait_async0();
#endif
    __syncthreads();
    for (int k = 2; k <= NPAD; k <<= 1) {
        for (int j = k >> 1; j > 0; j >>= 1) {
            for (int t = threadIdx.x; t < NPAD / 2; t += blockDim.x) {
                int i = 2 * t - (t & (j - 1));
                int ixj = i | j;
                unsigned long long a = skey[i], b = skey[ixj];
                bool swp = ((i & k) == 0) ? (a > b) : (a < b);
                if (swp) { skey[i] = b; skey[ixj] = a; }
            }
            __syncthreads();
        }
    }
    for (int i = threadIdx.x; i < NPAD; i += blockDim.x) keys[i] = skey[i];
}

// ---------------- 3) gather boxes into sorted order ----------------
__global__ void gather_kernel(const unsigned long long* __restrict__ keys,
                              const float4* __restrict__ box,
                              float4* __restrict__ sbox, unsigned* __restrict__ sord) {
    int r = blockIdx.x * blockDim.x + threadIdx.x;
    if (r < NBOX) {
        unsigned ord = (unsigned)(keys[r] & 0xFFFFFFFFu);
        sbox[r] = box[ord];
        sord[r] = ord;
    }
}

// ---------------- 4a) parallel suppression-mask build ----------------
// grid (53 col-tiles, 53 row-tiles), 256 threads; column boxes async-staged in LDS
__global__ void mask_kernel(const float4* __restrict__ sbox, unsigned* __restrict__ mask) {
    __shared__ float4 cbox[256];
    int c0 = blockIdx.x * 256;
    int cj = c0 + threadIdx.x;
    if (cj < NBOX) async_ld_b128(&cbox[threadIdx.x], &sbox[cj]);
    wait_async0();
    __syncthreads();
    int r = blockIdx.y * 256 + threadIdx.x;
    if (r >= NBOX) return;
    float4 rb = sbox[r];
    unsigned* row = mask + (size_t)r * NWPAD + (size_t)blockIdx.x * 8;
#pragma unroll
    for (int w = 0; w < 8; ++w) {
        unsigned m = 0;
        for (int b = 0; b < 32; ++b) {
            int j = c0 + w * 32 + b;
            if (j < NBOX && j > r) {
                if (iou(rb, cbox[w * 32 + b]) > IOU_TH) m |= (1u << b);
            }
        }
        row[w] = m;
    }
}

// ---------------- 4b) sequential greedy reduce (exact reference semantics) ---
__global__ void reduce_kernel(const unsigned* __restrict__ mask, unsigned* __restrict__ remv_out) {
    __shared__ unsigned remv[NWPAD];
    for (int w = threadIdx.x; w < NWPAD; w += blockDim.x) remv[w] = 0u;
    __syncthreads();
    for (int i = 0; i < NBOX; ++i) {
        unsigned word = remv[i >> 5];                  // uniform broadcast read
        bool active = ((word >> (i & 31)) & 1u) == 0u;
        if (active) {
            const unsigned* row = mask + (size_t)i * NWPAD;
            for (int w = threadIdx.x; w < NWPAD; w += blockDim.x) remv[w] |= row[w];
        }
        __syncthreads();
    }
    for (int w = threadIdx.x; w < NWPAD; w += blockDim.x) remv_out[w] = remv[w];
}

// ---------------- 4-fallback) whole NMS on one WGP, boxes resident in LDS ----
// dynamic LDS: 13520*16 B boxes + 424*4 B bitmask = ~218 KB (fits CDNA5 320 KB)
__global__ void nms_block_kernel(const float4* __restrict__ sbox, unsigned* __restrict__ remv_out) {
    float4* sb = (float4*)dynsmem;
    unsigned* remv = (unsigned*)(dynsmem + (size_t)NBOX * sizeof(float4));
    for (int i = threadIdx.x; i < NBOX; i += blockDim.x)
        async_ld_b128(&sb[i], &sbox[i]);
    for (int w = threadIdx.x; w < NWPAD; w += blockDim.x) remv[w] = 0u;
    wait_async0();
    __syncthreads();
    for (int i = 0; i < NBOX; ++i) {
        unsigned word = remv[i >> 5];
        bool active = ((word >> (i & 31)) & 1u) == 0u;
        if (active) {
            float4 bi = sb[i];
            for (int j = i + 1 + (int)threadIdx.x; j < NBOX; j += blockDim.x)
                if (iou(bi, sb[j]) > IOU_TH) atomicOr(&remv[j >> 5], 1u << (j & 31));
        }
        __syncthreads();
    }
    for (int w = threadIdx.x; w < NWPAD; w += blockDim.x) remv_out[w] = remv[w];
}

// ---------------- 5) scatter kept scores back to original slots -------------
__global__ void score_kernel(const unsigned* __restrict__ remv, const unsigned* __restrict__ sord,
                             const float* __restrict__ score, float* __restrict__ out) {
    int r = blockIdx.x * blockDim.x + threadIdx.x;
    if (r < NBOX) {
        bool keep = ((remv[r >> 5] >> (r & 31)) & 1u) == 0u;
        unsigned ord = sord[r];
        out[ord * 5 + 4] = keep ? score[ord] : 0.f;
    }
}

extern "C" void kernel_launch(void* const* d_in, const int* in_sizes, int n_in,
                              void* d_out, int out_size, void* d_ws, size_t ws_size,
                              hipStream_t stream) {
    (void)in_sizes; (void)n_in; (void)out_size;
    const float* x = (const float*)d_in[0];
    float* out = (float*)d_out;

    unsigned char* ws = (unsigned char*)d_ws;
    size_t off = 0;
    auto alloc = [&](size_t bytes) -> void* {
        void* p = ws + off;
        off += (bytes + 255) & ~(size_t)255;
        return p;
    };
    unsigned long long* keys = (unsigned long long*)alloc((size_t)NPAD * 8);
    float4*   box   = (float4*)alloc((size_t)NBOX * 16);
    float*    score = (float*)alloc((size_t)NBOX * 4);
    float4*   sbox  = (float4*)alloc((size_t)NBOX * 16);
    unsigned* sord  = (unsigned*)alloc((size_t)NBOX * 4);
    unsigned* remv  = (unsigned*)alloc((size_t)NWPAD * 4);
    unsigned* mask  = (unsigned*)alloc((size_t)NBOX * NWPAD * 4);  // ~23 MB
    bool use_mask = (ws_size >= off);

    // Allow >64KB dynamic LDS for the big-LDS kernels (no-op if unsupported;
    // not a stream op, deterministic, safe under graph capture).
    (void)hipFuncSetAttribute((const void*)sort_kernel,
                              hipFuncAttributeMaxDynamicSharedMemorySize,
                              NPAD * 8);
    (void)hipFuncSetAttribute((const void*)nms_block_kernel,
                              hipFuncAttributeMaxDynamicSharedMemorySize,
                              NBOX * 16 + NWPAD * 4);

    decode_kernel<<<(NPAD + 255) / 256, 256, 0, stream>>>(x, out, box, score, keys);
    sort_kernel<<<1, 1024, (size_t)NPAD * 8, stream>>>(keys);
    gather_kernel<<<(NBOX + 255) / 256, 256, 0, stream>>>(keys, box, sbox, sord);
    if (use_mask) {
        dim3 g((NBOX + 255) / 256, (NBOX + 255) / 256);   // 53 x 53
        mask_kernel<<<g, 256, 0, stream>>>(sbox, mask);
        reduce_kernel<<<1, 512, 0, stream>>>(mask, remv);
    } else {
        nms_block_kernel<<<1, 1024, (size_t)NBOX * 16 + (size_t)NWPAD * 4, stream>>>(sbox, remv);
    }
    score_kernel<<<(NBOX + 255) / 256, 256, 0, stream>>>(remv, sord, score, out);
}